// JEPAImpactModel_20736102105550
// MI455X (gfx1250) — compile-verified
//
#include <hip/hip_runtime.h>

#define T_LEN   16384
#define LATENT  16
#define BELIEF  32
#define IMP_HID 16
#define GI      96   // 3*BELIEF

typedef float v2f __attribute__((ext_vector_type(2)));
typedef float v8f __attribute__((ext_vector_type(8)));

// ---------------------------------------------------------------------------
// Fast activations using CDNA5 hardware transcendentals (V_TANH_F32,
// V_EXP_F32 (2^x), V_RCP_F32) -- straight-line TRANS ops on the serial path.
// ---------------------------------------------------------------------------
__device__ __forceinline__ float fast_tanh(float x) {
#if __has_builtin(__builtin_amdgcn_tanhf)
  return __builtin_amdgcn_tanhf(x);
#else
  return tanhf(x);
#endif
}

__device__ __forceinline__ float fast_sigmoid(float x) {
#if __has_builtin(__builtin_amdgcn_exp2f) && __has_builtin(__builtin_amdgcn_rcpf)
  // sigmoid(x) = 1 / (1 + 2^(-x*log2(e)))
  return __builtin_amdgcn_rcpf(1.0f + __builtin_amdgcn_exp2f(-1.4426950408889634f * x));
#else
  return 1.0f / (1.0f + expf(-x));
#endif
}

// Broadcast a float from a compile-time-constant lane: v_readlane_b32 -> SGPR,
// usable directly as the scalar operand of v_fmac (no ds_bpermute round-trip).
__device__ __forceinline__ float lane_bcast(float v, int lane) {
#if __has_builtin(__builtin_amdgcn_readlane)
  return __uint_as_float(__builtin_amdgcn_readlane(__float_as_uint(v), lane));
#else
  return __shfl(v, lane, 32);
#endif
}

// ---------------------------------------------------------------------------
// Kernel A: time-parallel precompute of gi_all[t][0:96] =
//   tanh(obs[t]*enc_w + enc_b) @ w_ih[0:16,:] + a_prev[t]*w_ih[16,:] + b_ih
// [T x 16] @ [16 x 96] fp32 GEMM via V_WMMA_F32_16X16X4_F32.
// One wave per (16-row time tile, 16-col output tile); K=16 in 4 chunks.
// ---------------------------------------------------------------------------
__global__ void __launch_bounds__(32)
gi_precompute_wmma(const float* __restrict__ obs,
                   const float* __restrict__ act,
                   const float* __restrict__ enc_w,
                   const float* __restrict__ enc_b,
                   const float* __restrict__ w_ih,
                   const float* __restrict__ b_ih,
                   float* __restrict__ gi_all) {
  const int lane = threadIdx.x;        // 0..31, full wave active (WMMA needs EXEC=all 1s)
  const int t0   = blockIdx.x * 16;    // time tile
  const int col0 = blockIdx.y * 16;    // output-column tile (6 of them)
  const int m    = lane & 15;          // A-matrix row index for this lane
  const int kh   = lane >> 4;          // half-wave selector for K / M split
  const int n    = m;                  // B/D column for this lane

  const float o = obs[t0 + m];         // OBS == 1

  v8f c = {0.f, 0.f, 0.f, 0.f, 0.f, 0.f, 0.f, 0.f};
#pragma unroll
  for (int kb = 0; kb < LATENT; kb += 4) {
    const int k0 = kb + 2 * kh;        // lanes 0-15: K=kb..kb+1, lanes 16-31: K=kb+2..kb+3
    v2f a, b;
    a[0] = fast_tanh(o * enc_w[k0]     + enc_b[k0]);      // z[t0+m][k0]
    a[1] = fast_tanh(o * enc_w[k0 + 1] + enc_b[k0 + 1]);  // z[t0+m][k0+1]
    b[0] = w_ih[(k0)     * GI + col0 + n];
    b[1] = w_ih[(k0 + 1) * GI + col0 + n];
    // D = A(16x4) x B(4x16) + C, fp32
    c = __builtin_amdgcn_wmma_f32_16x16x4_f32(false, a, false, b,
                                              (short)0, c, false, false);
  }

  const float wa = w_ih[LATENT * GI + col0 + n];  // action row (k = 16)
  const float bb = b_ih[col0 + n];
#pragma unroll
  for (int v = 0; v < 8; ++v) {
    const int t  = t0 + v + 8 * kh;                 // D layout: M = v + 8*half
    const int tm = (t > 0) ? (t - 1) : 0;           // clamped: load is always legal
    float ap = act[tm];
    ap = (t > 0) ? ap : 0.0f;                       // branch-free v_cndmask
    gi_all[t * GI + col0 + n] = c[v] + ap * wa + bb;
  }
}

// ---------------------------------------------------------------------------
// Kernel B: the serial scan. Single wave32; belief[j] lives in lane j.
// Weights register-resident; per-step inner loop = v_readlane + 4 FMAs.
// ---------------------------------------------------------------------------
__global__ void __launch_bounds__(32)
belief_scan(const float* __restrict__ gi_all,
            const float* __restrict__ act,
            const float* __restrict__ bounce,
            const float* __restrict__ w_hh,
            const float* __restrict__ b_hh,
            const float* __restrict__ imp_w1,
            const float* __restrict__ imp_b1,
            const float* __restrict__ imp_w2,
            const float* __restrict__ imp_b2,
            float* __restrict__ out) {
  const int j  = threadIdx.x;         // 0..31 : belief element owned by this lane
  const int jc = j & (IMP_HID - 1);   // impact hidden column (lanes 16-31 mirror 0-15)

  // ---- hoist all weights into registers (145 floats / lane + 5 biases) ----
  float whr[BELIEF], whz[BELIEF], whn[BELIEF];   // w_hh columns j, j+32, j+64
#pragma unroll
  for (int i = 0; i < BELIEF; ++i) {
    whr[i] = w_hh[i * GI + j];
    whz[i] = w_hh[i * GI + BELIEF + j];
    whn[i] = w_hh[i * GI + 2 * BELIEF + j];
  }
  float w1c[BELIEF + 1];                          // imp_w1 column jc (33 rows)
#pragma unroll
  for (int i = 0; i <= BELIEF; ++i) w1c[i] = imp_w1[i * IMP_HID + jc];
  float w2c[IMP_HID];                             // imp_w2 column j (16 rows)
#pragma unroll
  for (int k = 0; k < IMP_HID; ++k) w2c[k] = imp_w2[k * BELIEF + j];

  const float bhr = b_hh[j];
  const float bhz = b_hh[BELIEF + j];
  const float bhn = b_hh[2 * BELIEF + j];
  const float b1c = imp_b1[jc];
  const float b2c = imp_b2[j];

  float b = 0.0f;                     // belief[j]

  for (int t = 0; t < T_LEN; ++t) {
    // gh = belief @ w_hh + b_hh  (lane j owns columns j, j+32, j+64)
    float ghr  = bhr;
    float ghz  = bhz;
    float ghn  = bhn;
    float acc1 = b1c;                 // impact hidden pre-act (pre-update belief)
#pragma unroll
    for (int i = 0; i < BELIEF; ++i) {
      const float bi = lane_bcast(b, i);          // v_readlane_b32 -> SGPR
      ghr  += bi * whr[i];
      ghz  += bi * whz[i];
      ghn  += bi * whn[i];
      acc1 += bi * w1c[i];
    }

    const float ir  = gi_all[t * GI + j];
    const float iz  = gi_all[t * GI + BELIEF + j];
    const float in_ = gi_all[t * GI + 2 * BELIEF + j];
    __builtin_prefetch(&gi_all[(t + 1 < T_LEN ? t + 1 : t) * GI + j], 0, 3);

    const float r  = fast_sigmoid(ir + ghr);
    const float zg = fast_sigmoid(iz + ghz);
    const float nn = fast_tanh(in_ + r * ghn);
    const float bn = (1.0f - zg) * nn + zg * b;     // belief_normal

    // impact branch: delta = tanh([b, a_t] @ w1 + b1) @ w2 + b2
    const float a_t = act[t];
    acc1 += a_t * w1c[BELIEF];
    const float h = fast_tanh(acc1);                // valid per 16-col mirror
    float delta = b2c;
#pragma unroll
    for (int k = 0; k < IMP_HID; ++k) {
      delta += lane_bcast(h, k) * w2c[k];           // v_readlane_b32 -> SGPR
    }
    const float bimp = b + delta;

    const bool bnc = (bounce[t] == 1.0f) && (t > 0);
    b = bnc ? bimp : bn;
    out[t * BELIEF + j] = b;
  }
}

// ---------------------------------------------------------------------------
extern "C" void kernel_launch(void* const* d_in, const int* in_sizes, int n_in,
                              void* d_out, int out_size, void* d_ws, size_t ws_size,
                              hipStream_t stream) {
  (void)in_sizes; (void)n_in; (void)out_size; (void)ws_size;
  const float* obs    = (const float*)d_in[0];
  const float* act    = (const float*)d_in[1];
  const float* bounce = (const float*)d_in[2];
  const float* enc_w  = (const float*)d_in[3];
  const float* enc_b  = (const float*)d_in[4];
  const float* w_ih   = (const float*)d_in[5];
  const float* b_ih   = (const float*)d_in[6];
  const float* w_hh   = (const float*)d_in[7];
  const float* b_hh   = (const float*)d_in[8];
  const float* imp_w1 = (const float*)d_in[9];
  const float* imp_b1 = (const float*)d_in[10];
  const float* imp_w2 = (const float*)d_in[11];
  const float* imp_b2 = (const float*)d_in[12];
  float* out    = (float*)d_out;
  float* gi_all = (float*)d_ws;   // T * 96 floats = 6 MB scratch (fits in L2)

  dim3 gridA(T_LEN / 16, GI / 16);   // 1024 time tiles x 6 column tiles
  gi_precompute_wmma<<<gridA, 32, 0, stream>>>(obs, act, enc_w, enc_b,
                                               w_ih, b_ih, gi_all);
  belief_scan<<<1, 32, 0, stream>>>(gi_all, act, bounce, w_hh, b_hh,
                                    imp_w1, imp_b1, imp_w2, imp_b2, out);
}